// PositionLossVal_8452495638693
// MI455X (gfx1250) — compile-verified
//
#include <hip/hip_runtime.h>
#include <stdint.h>

#define HW   262144   // 512*512
#define NB   2048     // number of partial-sum blocks (grid)
#define TPB  256      // threads per block (8 waves of 32)
#define PXB  512      // pixels per block (TPB * 2)

typedef __attribute__((ext_vector_type(2))) float v2f;
typedef __attribute__((ext_vector_type(8))) float v8f;

// ---------------------------------------------------------------------------
// Kernel 1: fused per-pixel loss + deterministic block reduction.
// Each block covers 512 consecutive flattened pixels (never straddles a batch
// since 262144 % 512 == 0). Each thread handles 2 pixels (lane, lane+32 within
// its wave's 64-pixel slice). Flow channels 0..4 are staged wave-locally into
// LDS by the CDNA5 async engine while offset loads go straight to VGPRs.
// ---------------------------------------------------------------------------
__global__ __launch_bounds__(TPB) void pos_loss_partial_kernel(
    const float* __restrict__ off, const float* __restrict__ flow,
    float* __restrict__ partial)
{
  __shared__ float sflow[8 * 320];   // per wave: 5 channels * 64 pixels
  __shared__ float sred[TPB];

  const int tid  = threadIdx.x;
  const int lane = tid & 31;
  const int wave = tid >> 5;

  const long long P0 = (long long)blockIdx.x * PXB;
  const int b       = (int)(P0 >> 18);        // batch index
  const int pixbase = (int)(P0 & (HW - 1));   // pixel offset within batch

  const float* fbase = flow + ((long long)b * 8)  * HW + pixbase;
  const float* obase = off  + ((long long)b * 18) * HW + pixbase;

  const int wpix = wave * 64;                 // wave's pixel slice in block

  // ---- async-stage flow channels 0..4 for this wave's 64 pixels ----------
  {
    // low 32 bits of a generic shared address == LDS byte offset
    uint32_t lbase = (uint32_t)(uintptr_t)(&sflow[wave * 320]);
#pragma unroll
    for (int c = 0; c < 5; ++c) {
#pragma unroll
      for (int k = 0; k < 2; ++k) {
        uint32_t laddr = lbase + (uint32_t)((c * 64 + k * 32 + lane) * 4);
        uint64_t gaddr = (uint64_t)(uintptr_t)
            (fbase + (long long)c * HW + (wpix + k * 32 + lane));
        asm volatile("global_load_async_to_lds_b32 %0, %1, off"
                     :: "v"(laddr), "v"(gaddr) : "memory");
      }
    }
  }
  asm volatile("s_wait_asynccnt 0" ::: "memory");   // wave-local data: no barrier needed

  const float* wf = &sflow[wave * 320];

  float tsum = 0.0f;
#pragma unroll
  for (int k = 0; k < 2; ++k) {
    const int t = k * 32 + lane;               // 0..63, matches staged layout

    // per-segment precompute (hoisted out of the 9-point loop)
    float u[4], v[4], invd[4], invs[4];
#pragma unroll
    for (int j = 0; j < 4; ++j) {
      u[j] = wf[j * 64 + t];                   // flow channel j
      v[j] = wf[(j + 1) * 64 + t];             // flow channel j+1
      float uuvv = fmaf(u[j], u[j], v[j] * v[j]);
      float sq   = sqrtf(uuvv);
      invs[j] = 1.0f / sq;                     // 1/sqrt(uu_vv)
      invd[j] = invs[j] * invs[j];             // ~1/uu_vv
    }

    float isum = 0.0f;
#pragma unroll
    for (int i = 0; i < 9; ++i) {
      const float x  = obase[(long long)i * HW + wpix + t];
      const float y  = obase[(long long)(i + 9) * HW + wpix + t];
      const float xx = x * x;
      const float d1 = sqrtf(fmaf(x, x, y * y));
      float m = 3.402823466e+38f;
#pragma unroll
      for (int j = 0; j < 4; ++j) {
        // reference: x0 = (u*x*x + u*v*y)/uu_vv   (note x*x, kept verbatim)
        const float x0     = u[j] * fmaf(v[j], y, xx) * invd[j];
        const bool  inside = (fminf(0.0f, u[j]) <= x0) &&
                             (x0 <= fmaxf(0.0f, u[j]));
        const float perp   = fabsf(fmaf(v[j], x, -(u[j] * y))) * invs[j];
        const float dx = x - u[j], dy = y - v[j];
        const float d2 = sqrtf(fmaf(dx, dx, dy * dy));
        const float ed = fminf(d1, d2);
        m = fminf(m, inside ? perp : ed);
      }
      isum += m;
    }
    tsum += isum * (1.0f / 9.0f);              // mean over offset_num
  }

  // deterministic fixed-pairing tree reduction
  sred[tid] = tsum;
  __syncthreads();
#pragma unroll
  for (int s = TPB / 2; s > 0; s >>= 1) {
    if (tid < s) sred[tid] += sred[tid + s];
    __syncthreads();
  }
  if (tid == 0) partial[blockIdx.x] = sred[0];
}

// ---------------------------------------------------------------------------
// Kernel 2: sum the 2048 partials exactly in f32 with WMMA:
//   D += A(16x4) x ones(4x16); 64 values per step, 32 steps, fixed order.
// A layout (f32 16x4): lanes 0-15 hold (m=lane, k=0,1) in vgpr0/1,
// lanes 16-31 hold (m=lane-16, k=2,3). Every column of D then holds the 16
// row-sums; fold with two lane reads (lane 0 -> m=0..7, lane 16 -> m=8..15).
// ---------------------------------------------------------------------------
__global__ __launch_bounds__(32) void pos_loss_final_kernel(
    const float* __restrict__ partial, float* __restrict__ out)
{
  const int lane = threadIdx.x;       // one full wave32, EXEC all ones
  const int m    = lane & 15;
  const int kk   = (lane >> 4) * 2;   // K base: 0 (lanes 0-15) or 2 (16-31)

  v8f c = {};
  v2f onesb; onesb.x = 1.0f; onesb.y = 1.0f;

#pragma unroll 4
  for (int ch = 0; ch < NB / 64; ++ch) {
    const int idx = ch * 64 + m * 4 + kk;   // A[m][k] = partial[ch*64+m*4+k]
    v2f a; a.x = partial[idx]; a.y = partial[idx + 1];
    c = __builtin_amdgcn_wmma_f32_16x16x4_f32(
        /*neg_a=*/false, a, /*neg_b=*/false, onesb,
        /*c_mod=*/(short)0, c, /*reuse_a=*/false, /*reuse_b=*/false);
  }

  float s = c[0] + c[1] + c[2] + c[3] + c[4] + c[5] + c[6] + c[7];
  const float tot = __shfl(s, 0, 32) + __shfl(s, 16, 32);
  if (lane == 0) out[0] = tot * (1.0f / 262144.0f);   // / (h*w), exact 2^-18
}

extern "C" void kernel_launch(void* const* d_in, const int* in_sizes, int n_in,
                              void* d_out, int out_size, void* d_ws, size_t ws_size,
                              hipStream_t stream) {
  (void)in_sizes; (void)n_in; (void)out_size; (void)ws_size;
  const float* off  = (const float*)d_in[0];   // [4,18,512,512] f32
  const float* flow = (const float*)d_in[1];   // [4, 8,512,512] f32
  float* partial = (float*)d_ws;               // 2048 f32 block partials
  float* outp    = (float*)d_out;              // scalar f32

  pos_loss_partial_kernel<<<NB, TPB, 0, stream>>>(off, flow, partial);
  pos_loss_final_kernel<<<1, 32, 0, stream>>>(partial, outp);
}